// AdaptiveGatingMetaNet_773094113737
// MI455X (gfx1250) — compile-verified
//
#include <hip/hip_runtime.h>

typedef __attribute__((ext_vector_type(2))) float v2f;
typedef __attribute__((ext_vector_type(8))) float v8f;

#define BM 128
#define BN 128
#define BK 16
#define PAD_A 20    // floats per A / B-transposed LDS row (80B: 16B aligned, bank-spread)
#define PAD_B 132   // floats per row-major-B LDS row (528B: 16B aligned)

static __device__ __forceinline__ v8f wmma_f32(v2f a, v2f b, v8f c) {
  // V_WMMA_F32_16X16X4_F32: D = A(16x4) * B(4x16) + C(16x16), fp32 matrix pipe
  return __builtin_amdgcn_wmma_f32_16x16x4_f32(false, a, false, b, (short)0, c,
                                               false, false);
}

// gfx1250 async DMA: global -> LDS, tracked by ASYNCcnt (no wave-side load wait,
// no ds_store issue slots). One instruction moves 16B per lane = 512B per wave.
static __device__ __forceinline__ void async_b128(unsigned lds_off, const float* g) {
  asm volatile("global_load_async_to_lds_b128 %0, %1, off"
               :: "v"(lds_off), "v"((unsigned long long)(size_t)g)
               : "memory");
}
static __device__ __forceinline__ void wait_async0() {
  asm volatile("s_wait_asynccnt 0x0" ::: "memory");
}

// ---------------------------------------------------------------------------
// GEMM with transposed B source: C[m,n] = sum_k A[m*lda+k] * Bsrc[n*ldb+k] (+bias[n])
// Used for: h = f @ W1^T (+b1), G = W1 @ W1^T, out = x @ projW^T
// ---------------------------------------------------------------------------
__global__ __launch_bounds__(256)
void gemm_bt_kernel(const float* __restrict__ A, int lda,
                    const float* __restrict__ Bsrc, int ldb,
                    float* __restrict__ C, int ldc,
                    const float* __restrict__ bias, int K)
{
  __shared__ __align__(16) float sA[2][BM][PAD_A];
  __shared__ __align__(16) float sBT[2][BN][PAD_A];

  const int tid  = threadIdx.x;
  const int lane = tid & 31;
  const int wave = tid >> 5;
  const int wm   = wave & 1;        // 2 waves along M (64 rows each)
  const int wn   = wave >> 1;       // 4 waves along N (32 cols each)
  const int row0 = blockIdx.x * BM;
  const int col0 = blockIdx.y * BN;

  // per-thread staging descriptors (2 x 16B chunks of A, 2 of B^T per stage)
  const float* ga[2];
  const float* gb[2];
  unsigned la[2], lb[2];
#pragma unroll
  for (int i = 0; i < 2; ++i) {
    int idx = tid + i * 256;           // 512 float4 per tile
    int r   = idx >> 2;                // 0..127
    int kq  = (idx & 3) << 2;          // 0,4,8,12
    ga[i] = A    + (size_t)(row0 + r) * lda + kq;
    gb[i] = Bsrc + (size_t)(col0 + r) * ldb + kq;
    la[i] = (unsigned)(size_t)&sA[0][r][kq];
    lb[i] = (unsigned)(size_t)&sBT[0][r][kq];
  }
  const unsigned dA = (unsigned)sizeof(float) * BM * PAD_A;
  const unsigned dB = (unsigned)sizeof(float) * BN * PAD_A;

  v8f acc[4][2];
#pragma unroll
  for (int mt = 0; mt < 4; ++mt)
#pragma unroll
    for (int nt = 0; nt < 2; ++nt) acc[mt][nt] = (v8f){0,0,0,0,0,0,0,0};

  // stage 0 -> buffer 0 via async DMA
#pragma unroll
  for (int i = 0; i < 2; ++i) { async_b128(la[i], ga[i]); async_b128(lb[i], gb[i]); }
  wait_async0();
  __syncthreads();

  const int nk    = K / BK;
  const int mrow  = lane & 15;
  const int khalf = (lane >> 4) << 1;  // lanes 16..31 hold K+2,K+3 (ISA A layout)

  for (int t = 0; t < nk; ++t) {
    const int buf = t & 1;
    if (t + 1 < nk) {                  // fire-and-forget DMA for next K-slice
      const unsigned nb = (unsigned)(buf ^ 1);
      const int k0 = (t + 1) * BK;
#pragma unroll
      for (int i = 0; i < 2; ++i) {
        async_b128(la[i] + nb * dA, ga[i] + k0);
        async_b128(lb[i] + nb * dB, gb[i] + k0);
      }
    }
#pragma unroll
    for (int kk = 0; kk < BK; kk += 4) {
      v2f a[4], b[2];
#pragma unroll
      for (int mt = 0; mt < 4; ++mt)
        a[mt] = *(const v2f*)&sA[buf][wm * 64 + mt * 16 + mrow][kk + khalf];
#pragma unroll
      for (int nt = 0; nt < 2; ++nt)
        b[nt] = *(const v2f*)&sBT[buf][wn * 32 + nt * 16 + mrow][kk + khalf];
#pragma unroll
      for (int mt = 0; mt < 4; ++mt)
#pragma unroll
        for (int nt = 0; nt < 2; ++nt)
          acc[mt][nt] = wmma_f32(a[mt], b[nt], acc[mt][nt]);
    }
    if (t + 1 < nk) {
      wait_async0();     // my DMA for stage t+1 landed
      __syncthreads();   // everyone's landed; everyone done reading buf
    }
  }

  const int mhalf = (lane >> 4) << 3;  // C/D layout: lanes 16..31 hold M = r+8
#pragma unroll
  for (int mt = 0; mt < 4; ++mt) {
    const int mbase = row0 + wm * 64 + mt * 16 + mhalf;
#pragma unroll
    for (int nt = 0; nt < 2; ++nt) {
      const int nn = col0 + wn * 32 + nt * 16 + (lane & 15);
      const float bv = bias ? bias[nn] : 0.0f;
      v8f c = acc[mt][nt];
#pragma unroll
      for (int r = 0; r < 8; ++r)
        C[(size_t)(mbase + r) * ldc + nn] = c[r] + bv;
    }
  }
}

// ---------------------------------------------------------------------------
// Chain step: Y = X + diag(gated[:,j]) * (X @ T),  T row-major [D,D]
// ---------------------------------------------------------------------------
__global__ __launch_bounds__(256)
void gemm_chain_kernel(const float* __restrict__ X,
                       const float* __restrict__ T,
                       const float* __restrict__ gated, int j,
                       float* __restrict__ Y, int D)
{
  __shared__ __align__(16) float sA[2][BM][PAD_A];
  __shared__ __align__(16) float sB[2][BK][PAD_B];

  const int tid  = threadIdx.x;
  const int lane = tid & 31;
  const int wave = tid >> 5;
  const int wm   = wave & 1;
  const int wn   = wave >> 1;
  const int row0 = blockIdx.x * BM;
  const int col0 = blockIdx.y * BN;

  const float* ga[2];
  const float* gb[2];
  unsigned la[2], lb[2];
#pragma unroll
  for (int i = 0; i < 2; ++i) {
    int idx = tid + i * 256;
    int r   = idx >> 2,  kq = (idx & 3) << 2;    // A: 128 rows x 16 k
    int kr  = idx >> 5,  nc = (idx & 31) << 2;   // B: 16 k-rows x 128 cols
    ga[i] = X + (size_t)(row0 + r) * D + kq;
    gb[i] = T + (size_t)kr * D + col0 + nc;
    la[i] = (unsigned)(size_t)&sA[0][r][kq];
    lb[i] = (unsigned)(size_t)&sB[0][kr][nc];
  }
  const unsigned dA = (unsigned)sizeof(float) * BM * PAD_A;
  const unsigned dB = (unsigned)sizeof(float) * BK * PAD_B;
  const int bstep = BK * D;                       // B advances 16 rows per stage

  v8f acc[4][2];
#pragma unroll
  for (int mt = 0; mt < 4; ++mt)
#pragma unroll
    for (int nt = 0; nt < 2; ++nt) acc[mt][nt] = (v8f){0,0,0,0,0,0,0,0};

#pragma unroll
  for (int i = 0; i < 2; ++i) { async_b128(la[i], ga[i]); async_b128(lb[i], gb[i]); }
  wait_async0();
  __syncthreads();

  const int nk    = D / BK;
  const int mrow  = lane & 15;
  const int khalf = (lane >> 4) << 1;

  for (int t = 0; t < nk; ++t) {
    const int buf = t & 1;
    if (t + 1 < nk) {
      const unsigned nb = (unsigned)(buf ^ 1);
      const int k0 = (t + 1) * BK;
#pragma unroll
      for (int i = 0; i < 2; ++i) {
        async_b128(la[i] + nb * dA, ga[i] + k0);
        async_b128(lb[i] + nb * dB, gb[i] + (size_t)(t + 1) * bstep);
      }
    }
#pragma unroll
    for (int kk = 0; kk < BK; kk += 4) {
      v2f a[4], b[2];
#pragma unroll
      for (int mt = 0; mt < 4; ++mt)
        a[mt] = *(const v2f*)&sA[buf][wm * 64 + mt * 16 + mrow][kk + khalf];
#pragma unroll
      for (int nt = 0; nt < 2; ++nt) {
        int n = wn * 32 + nt * 16 + mrow;
        v2f bb;
        bb.x = sB[buf][kk + khalf][n];
        bb.y = sB[buf][kk + khalf + 1][n];
        b[nt] = bb;
      }
#pragma unroll
      for (int mt = 0; mt < 4; ++mt)
#pragma unroll
        for (int nt = 0; nt < 2; ++nt)
          acc[mt][nt] = wmma_f32(a[mt], b[nt], acc[mt][nt]);
    }
    if (t + 1 < nk) {
      wait_async0();
      __syncthreads();
    }
  }

  const int mhalf = (lane >> 4) << 3;
#pragma unroll
  for (int mt = 0; mt < 4; ++mt) {
    const int mbase = row0 + wm * 64 + mt * 16 + mhalf;
#pragma unroll
    for (int nt = 0; nt < 2; ++nt) {
      const int nn = col0 + wn * 32 + nt * 16 + (lane & 15);
      v8f c = acc[mt][nt];
#pragma unroll
      for (int r = 0; r < 8; ++r) {
        const int row = mbase + r;
        float g = gated[(size_t)row * 8 + j];
        Y[(size_t)row * D + nn] = X[(size_t)row * D + nn] + g * c[r];
      }
    }
  }
}

// ---------------------------------------------------------------------------
// unc²[b,k] = q G qᵀ,  q[b,h] = (h[b,h]>0) ? W2[k,h] : 0.  A built on the fly
// (manual staging), B (=G) staged via async DMA. Fused epilogue dots S=qG
// against q and atomically accumulates into unc2[b,k]. blockIdx.z = task k.
// ---------------------------------------------------------------------------
__global__ __launch_bounds__(256)
void gemm_unc_kernel(const float* __restrict__ hbuf,
                     const float* __restrict__ W2,
                     const float* __restrict__ G,
                     float* __restrict__ unc2, int H)
{
  __shared__ __align__(16) float sA[2][BM][PAD_A];
  __shared__ __align__(16) float sB[2][BK][PAD_B];

  const int tid  = threadIdx.x;
  const int lane = tid & 31;
  const int wave = tid >> 5;
  const int wm   = wave & 1;
  const int wn   = wave >> 1;
  const int row0 = blockIdx.x * BM;
  const int col0 = blockIdx.y * BN;
  const int kidx = blockIdx.z;

  const float* gh[2];
  const float* gw[2];
  const float* gb[2];
  float* lsa[2];
  unsigned lb[2];
#pragma unroll
  for (int i = 0; i < 2; ++i) {
    int idx = tid + i * 256;
    int r   = idx >> 2,  kq = (idx & 3) << 2;
    int kr  = idx >> 5,  nc = (idx & 31) << 2;
    gh[i]  = hbuf + (size_t)(row0 + r) * H + kq;
    gw[i]  = W2 + (size_t)kidx * H + kq;
    gb[i]  = G + (size_t)kr * H + col0 + nc;
    lsa[i] = &sA[0][r][kq];
    lb[i]  = (unsigned)(size_t)&sB[0][kr][nc];
  }
  const unsigned dA = (unsigned)sizeof(float) * BM * PAD_A;  // in floats below
  const unsigned dB = (unsigned)sizeof(float) * BK * PAD_B;
  const int aStrideF = BM * PAD_A;     // float stride between sA buffers
  const int bstep = BK * H;

  v8f acc[4][2];
#pragma unroll
  for (int mt = 0; mt < 4; ++mt)
#pragma unroll
    for (int nt = 0; nt < 2; ++nt) acc[mt][nt] = (v8f){0,0,0,0,0,0,0,0};

  // stage 0: async B, manual masked A
#pragma unroll
  for (int i = 0; i < 2; ++i) {
    async_b128(lb[i], gb[i]);
    float4 hv = *(const float4*)gh[i];
    float4 wv = *(const float4*)gw[i];
    float4 q;
    q.x = hv.x > 0.0f ? wv.x : 0.0f;
    q.y = hv.y > 0.0f ? wv.y : 0.0f;
    q.z = hv.z > 0.0f ? wv.z : 0.0f;
    q.w = hv.w > 0.0f ? wv.w : 0.0f;
    *(float4*)lsa[i] = q;
  }
  wait_async0();
  __syncthreads();

  const int nk    = H / BK;
  const int mrow  = lane & 15;
  const int khalf = (lane >> 4) << 1;

  for (int t = 0; t < nk; ++t) {
    const int buf = t & 1;
    float4 ph[2], pw[2];
    if (t + 1 < nk) {
      const unsigned nb = (unsigned)(buf ^ 1);
      const int k0 = (t + 1) * BK;
#pragma unroll
      for (int i = 0; i < 2; ++i) {
        async_b128(lb[i] + nb * dB, gb[i] + (size_t)(t + 1) * bstep);
        ph[i] = *(const float4*)(gh[i] + k0);
        pw[i] = *(const float4*)(gw[i] + k0);
      }
    }
#pragma unroll
    for (int kk = 0; kk < BK; kk += 4) {
      v2f a[4], b[2];
#pragma unroll
      for (int mt = 0; mt < 4; ++mt)
        a[mt] = *(const v2f*)&sA[buf][wm * 64 + mt * 16 + mrow][kk + khalf];
#pragma unroll
      for (int nt = 0; nt < 2; ++nt) {
        int n = wn * 32 + nt * 16 + mrow;
        v2f bb;
        bb.x = sB[buf][kk + khalf][n];
        bb.y = sB[buf][kk + khalf + 1][n];
        b[nt] = bb;
      }
#pragma unroll
      for (int mt = 0; mt < 4; ++mt)
#pragma unroll
        for (int nt = 0; nt < 2; ++nt)
          acc[mt][nt] = wmma_f32(a[mt], b[nt], acc[mt][nt]);
    }
    if (t + 1 < nk) {
      const int nb = buf ^ 1;
#pragma unroll
      for (int i = 0; i < 2; ++i) {
        float4 q;
        q.x = ph[i].x > 0.0f ? pw[i].x : 0.0f;
        q.y = ph[i].y > 0.0f ? pw[i].y : 0.0f;
        q.z = ph[i].z > 0.0f ? pw[i].z : 0.0f;
        q.w = ph[i].w > 0.0f ? pw[i].w : 0.0f;
        *(float4*)(lsa[i] + nb * aStrideF) = q;
      }
      wait_async0();
      __syncthreads();
    }
  }

  // Fused epilogue: p = sum_n S[b,n]*q[b,n] over this block's 128 columns,
  // cross-lane reduce within each 16-lane half, atomic accumulate.
  const int mhalf = (lane >> 4) << 3;
#pragma unroll
  for (int mt = 0; mt < 4; ++mt) {
    const int mbase = row0 + wm * 64 + mt * 16 + mhalf;
#pragma unroll
    for (int r = 0; r < 8; ++r) {
      const int row = mbase + r;
      float p = 0.0f;
#pragma unroll
      for (int nt = 0; nt < 2; ++nt) {
        const int nn = col0 + wn * 32 + nt * 16 + (lane & 15);
        float hv = hbuf[(size_t)row * H + nn];
        float q  = hv > 0.0f ? W2[(size_t)kidx * H + nn] : 0.0f;
        p += acc[mt][nt][r] * q;
      }
      p += __shfl_xor(p, 1);
      p += __shfl_xor(p, 2);
      p += __shfl_xor(p, 4);
      p += __shfl_xor(p, 8);
      if ((lane & 15) == 0) atomicAdd(&unc2[(size_t)row * 8 + kidx], p);
    }
  }
}

// ---------------------------------------------------------------------------
// Small helper kernels
// ---------------------------------------------------------------------------
__global__ void zero_kernel(float* unc2, unsigned* mxbits, int n) {
  int i = blockIdx.x * blockDim.x + threadIdx.x;
  if (i < n) unc2[i] = 0.0f;
  if (i == 0) *mxbits = 0u;
}

__global__ void coeffs_kernel(const float* __restrict__ h, const float* __restrict__ W2,
                              const float* __restrict__ b2, float* __restrict__ coeffs,
                              int n, int H) {
  int i = blockIdx.x * blockDim.x + threadIdx.x;   // i = b*8 + k
  if (i >= n) return;
  int b = i >> 3, k = i & 7;
  const float* hr = h + (size_t)b * H;
  const float* wr = W2 + (size_t)k * H;
  float s = b2[k];
  for (int t = 0; t < H; ++t) s = fmaf(fmaxf(hr[t], 0.0f), wr[t], s);
  coeffs[i] = s;
}

__global__ void maxred_kernel(const float* __restrict__ unc2, unsigned* mxbits, int n) {
  int i = blockIdx.x * blockDim.x + threadIdx.x;
  if (i < n) {
    float u = sqrtf(fmaxf(unc2[i], 0.0f));
    atomicMax(mxbits, __float_as_uint(u));  // valid: u >= 0
  }
}

__global__ void gate_kernel(const float* __restrict__ coeffs, const float* __restrict__ unc2,
                            const unsigned* __restrict__ mxbits,
                            const float* __restrict__ base_t, const float* __restrict__ beta,
                            float* __restrict__ gated, int n) {
  int i = blockIdx.x * blockDim.x + threadIdx.x;
  if (i >= n) return;
  float mx = __uint_as_float(*mxbits);
  float u  = sqrtf(fmaxf(unc2[i], 0.0f));
  if (mx > 0.0f) u /= mx;
  float bt   = base_t[0];
  float base = (bt > 20.0f) ? bt : log1pf(expf(bt));   // softplus
  float thr  = base * (1.0f + fmaxf(beta[0], 0.0f) * u);
  float c    = coeffs[i];
  gated[i]   = (fabsf(c) < thr) ? 0.0f : c;
}

// ---------------------------------------------------------------------------
extern "C" void kernel_launch(void* const* d_in, const int* in_sizes, int n_in,
                              void* d_out, int out_size, void* d_ws, size_t ws_size,
                              hipStream_t stream) {
  const float* features  = (const float*)d_in[0];
  const float* W1        = (const float*)d_in[1];
  const float* b1        = (const float*)d_in[2];
  const float* W2        = (const float*)d_in[3];
  const float* b2        = (const float*)d_in[4];
  const float* task_mats = (const float*)d_in[5];
  const float* proj_W    = (const float*)d_in[6];
  const float* base_t    = (const float*)d_in[7];
  const float* beta      = (const float*)d_in[8];

  const int B = 8192, D = 1024, H = 256, NT = 8;

  float* ws = (float*)d_ws;
  float* h        = ws;            ws += (size_t)B * H;   // 8 MB
  float* G        = ws;            ws += (size_t)H * H;
  float* coeffs   = ws;            ws += (size_t)B * NT;
  float* unc2     = ws;            ws += (size_t)B * NT;
  unsigned* mxbits = (unsigned*)ws; ws += 64;
  float* gated    = ws;            ws += (size_t)B * NT;
  float* xB       = ws;            ws += (size_t)B * D;   // 32 MB
  float* xA       = (float*)d_out;                        // reuse output as ping buffer

  dim3 blk(256);
  int nSmall = B * NT;

  zero_kernel<<<(nSmall + 255) / 256, blk, 0, stream>>>(unc2, mxbits, nSmall);

  // h = features @ W1^T + b1
  gemm_bt_kernel<<<dim3(B / BM, H / BN), blk, 0, stream>>>(features, D, W1, D, h, H, b1, D);

  // coeffs = relu(h) @ W2^T + b2
  coeffs_kernel<<<(nSmall + 255) / 256, blk, 0, stream>>>(h, W2, b2, coeffs, nSmall, H);

  // G = W1 @ W1^T
  gemm_bt_kernel<<<dim3(H / BM, H / BN), blk, 0, stream>>>(W1, D, W1, D, G, H, nullptr, D);

  // unc2[b,k] = (mask_b * W2_k) G (mask_b * W2_k)^T
  gemm_unc_kernel<<<dim3(B / BM, H / BN, NT), blk, 0, stream>>>(h, W2, G, unc2, H);

  // global max and gating
  maxred_kernel<<<(nSmall + 255) / 256, blk, 0, stream>>>(unc2, mxbits, nSmall);
  gate_kernel<<<(nSmall + 255) / 256, blk, 0, stream>>>(coeffs, unc2, mxbits, base_t, beta,
                                                        gated, nSmall);

  // sequential task-vector chain: x = x + gated[:,j] * (x @ T_j)
  const float* xin = features;
  for (int j = 0; j < NT; ++j) {
    float* xout = (j & 1) ? xB : xA;           // j=7 ends in xB
    gemm_chain_kernel<<<dim3(B / BM, D / BN), blk, 0, stream>>>(
        xin, task_mats + (size_t)j * D * D, gated, j, xout, D);
    xin = xout;
  }

  // out = x @ proj_W^T   (reads xB, writes d_out)
  gemm_bt_kernel<<<dim3(B / BM, D / BN), blk, 0, stream>>>(xB, D, proj_W, D, (float*)d_out, D,
                                                           nullptr, D);
}